// GraphProcessor_64012192579962
// MI455X (gfx1250) — compile-verified
//
#include <hip/hip_runtime.h>
#include <stdint.h>

#define CUTOFF   5.0f
#define TILE     256                 // edges per block-tile
#define NTHREADS 256                 // 8 wave32 waves
#define COPIERS  (TILE * 12 / 16)    // 192 threads copy 16B each -> 3072B tile

__global__ __launch_bounds__(NTHREADS)
void edge_switch_kernel(const float* __restrict__ coords,   // [N,3]
                        const int*   __restrict__ esrc,     // [E]
                        const int*   __restrict__ edst,     // [E]
                        const float* __restrict__ pbc,      // [E,3]
                        const float* __restrict__ cells,    // [1,3,3]
                        float*       __restrict__ out,      // vec(3E)|dist(E)|sw(E)|mask(E)
                        int E)
{
    __shared__ __align__(16) float sbuf[2][TILE * 3];  // double-buffered pbc tiles
    __shared__ float scell[9];

    const int tid      = threadIdx.x;
    const int numTiles = (E + TILE - 1) / TILE;
    const int stride   = (int)gridDim.x;
    const size_t Ebytes = (size_t)E * 12u;

    if (tid < 9) scell[tid] = cells[tid];

    // ---- prologue: async-stage first tile of this block into buffer 0 ----
    int t0 = (int)blockIdx.x;
    if (t0 < numTiles && tid < COPIERS) {
        size_t byteOff = (size_t)t0 * (TILE * 12) + (size_t)tid * 16;
        if (byteOff < Ebytes) {
            const char* g = (const char*)pbc + byteOff;
            uint32_t l = (uint32_t)(uintptr_t)((const char*)&sbuf[0][0] + tid * 16);
            asm volatile("global_load_async_to_lds_b128 %0, %1, off"
                         :: "v"(l), "v"(g) : "memory");
        }
    }

    int buf = 0;
    for (int t = t0; t < numTiles; t += stride, buf ^= 1) {
        const int  tn      = t + stride;
        const bool hasNext = tn < numTiles;

        // ---- stage NEXT tile into the other buffer (async, overlapped) ----
        if (hasNext && tid < COPIERS) {
            size_t byteOff = (size_t)tn * (TILE * 12) + (size_t)tid * 16;
            if (byteOff < Ebytes) {
                const char* g = (const char*)pbc + byteOff;
                uint32_t l = (uint32_t)(uintptr_t)((const char*)&sbuf[buf ^ 1][0] + tid * 16);
                asm volatile("global_load_async_to_lds_b128 %0, %1, off"
                             :: "v"(l), "v"(g) : "memory");
            }
        }
        // prefetch next tile's indices while we're at it
        if (hasNext) {
            int ni = tn * TILE + tid;
            if (ni < E) {
                __builtin_prefetch(esrc + ni, 0, 0);
                __builtin_prefetch(edst + ni, 0, 0);
            }
        }

        // ---- wait: current tile staged (<=1 leaves next stage in flight) ----
        if (tid < COPIERS) {
            if (hasNext) asm volatile("s_wait_asynccnt 0x1" ::: "memory");
            else         asm volatile("s_wait_asynccnt 0x0" ::: "memory");
        }
        __syncthreads();   // all waves see staged tile (and scell on 1st iter)

        // ---- compute ----
        const int i = t * TILE + tid;
        if (i < E) {
            const int s = esrc[i];
            const int d = edst[i];
            const float ax = coords[3 * s + 0], ay = coords[3 * s + 1], az = coords[3 * s + 2];
            const float bx = coords[3 * d + 0], by = coords[3 * d + 1], bz = coords[3 * d + 2];
            const float px = sbuf[buf][3 * tid + 0];
            const float py = sbuf[buf][3 * tid + 1];
            const float pz = sbuf[buf][3 * tid + 2];

            // vec = (dst - src) + p @ C   with (p@C)[j] = sum_k p[k]*C[k][j]
            const float vx = (bx - ax) + px * scell[0] + py * scell[3] + pz * scell[6];
            const float vy = (by - ay) + px * scell[1] + py * scell[4] + pz * scell[7];
            const float vz = (bz - az) + px * scell[2] + py * scell[5] + pz * scell[8];

            const float d2   = vx * vx + vy * vy + vz * vz;
            const float dist = __builtin_amdgcn_sqrtf(d2);           // v_sqrt_f32
            const bool  in   = dist < CUTOFF;
            // v_cos_f32(x) = cos(2*pi*x)  =>  cos(pi*d/CUTOFF) = v_cos(d * 0.5/CUTOFF)
            const float c    = __builtin_amdgcn_cosf(dist * (0.5f / CUTOFF));
            const float sw   = in ? (0.5f * c + 0.5f) : 0.0f;

            const size_t ii = (size_t)i;
            out[3 * ii + 0] = vx;
            out[3 * ii + 1] = vy;
            out[3 * ii + 2] = vz;
            const size_t base = (size_t)3 * (size_t)E;
            out[base + ii]                     = dist;            // distances
            out[base + (size_t)E + ii]         = sw;              // switch
            out[base + 2 * (size_t)E + ii]     = in ? 1.0f : 0.0f; // edge_mask
        }
        __syncthreads();   // everyone done reading sbuf[buf] before it is restaged
    }
}

extern "C" void kernel_launch(void* const* d_in, const int* in_sizes, int n_in,
                              void* d_out, int out_size, void* d_ws, size_t ws_size,
                              hipStream_t stream) {
    (void)n_in; (void)out_size; (void)d_ws; (void)ws_size;
    const float* coords = (const float*)d_in[0];
    const int*   esrc   = (const int*)d_in[1];
    const int*   edst   = (const int*)d_in[2];
    const float* pbc    = (const float*)d_in[3];
    const float* cells  = (const float*)d_in[4];
    const int E = in_sizes[1];                       // edge_src element count

    const int numTiles = (E + TILE - 1) / TILE;
    int blocks = numTiles < 2048 ? numTiles : 2048;  // grid-stride over tiles
    if (blocks < 1) blocks = 1;

    edge_switch_kernel<<<blocks, NTHREADS, 0, stream>>>(
        coords, esrc, edst, pbc, cells, (float*)d_out, E);
}